// EchoStateNetwork_23510650978631
// MI455X (gfx1250) — compile-verified
//
#include <hip/hip_runtime.h>

// ============================================================================
// Echo State Network on MI455X (gfx1250, wave32)
//
// Phase 1: pre = train_data[2048,512] @ w_in[512,4096]  (f32 GEMM)
//   -> native f32 WMMA (V_WMMA_F32_16X16X4_F32), one wave per 16x16 tile,
//      128 WMMA accumulations over K=512. Bit-compatible with FMA math.
//
// Phase 2: sequential scan x_t = tanh(pre_t / (x_{t-1} @ w_res))
//   -> w_res is 64MB f32: fits entirely in the 192MB global L2, so the scan
//      is an L2-bandwidth-bound streaming GEMV (131 GB of L2 traffic total).
//      Persistent kernel, 128 workgroups * 32 columns each (one 128B line per
//      row per WG), grid barrier per step via atomic counter, state vector
//      double-buffered in workspace + staged in LDS.
// ============================================================================

typedef float v2f __attribute__((ext_vector_type(2)));
typedef float v8f __attribute__((ext_vector_type(8)));

#define T_STEPS 2048
#define K_DIM   512
#define R_DIM   4096
#define NBLK    128     // persistent workgroups for the scan (<= WGP count)
#define TPB     256     // 8 waves of 32

// --------------------------------------------------------------------------
// Phase 1: f32 WMMA GEMM, pre -> out (rows later overwritten by x_t)
// One wave computes one 16x16 tile of out. ISA 7.12.2 layouts:
//   A 16x4 : lane l<16 holds A[m0+l][k..k+1], lane l>=16 holds A[m0+l-16][k+2..k+3]
//   B 4x16 : v0 = B[k][n0+lm] (lo lanes) / B[k+2][n0+lm] (hi lanes), v1 = k+1/k+3
//   C/D    : vgpr r = row m0 + r + 8*hi, col n0 + lm
// --------------------------------------------------------------------------
__global__ __launch_bounds__(TPB) void esn_pre_gemm(
    const float* __restrict__ A,   // [T_STEPS, K_DIM]
    const float* __restrict__ B,   // [K_DIM, R_DIM]
    float* __restrict__ out)       // [T_STEPS, R_DIM]
{
    const int wave = (blockIdx.x * TPB + threadIdx.x) >> 5;
    const int lane = threadIdx.x & 31;
    const int tiles_n = R_DIM / 16;            // 256
    const int m0 = (wave / tiles_n) * 16;
    const int n0 = (wave % tiles_n) * 16;
    const int lm = lane & 15;
    const int hi = lane >> 4;

    v8f c = {};
    const float* arow = A + (size_t)(m0 + lm) * K_DIM + 2 * hi;
    const float* bcol = B + (size_t)(2 * hi) * R_DIM + n0 + lm;

    for (int k = 0; k < K_DIM; k += 4) {
        v2f a;
        a.x = arow[k];
        a.y = arow[k + 1];
        v2f b;
        const float* bp = bcol + (size_t)k * R_DIM;
        b.x = bp[0];
        b.y = bp[R_DIM];
        // D = A(16x4) * B(4x16) + C, full-precision f32 matrix pipe
        c = __builtin_amdgcn_wmma_f32_16x16x4_f32(
                /*neg_a=*/false, a, /*neg_b=*/false, b,
                /*c_mod=*/(short)0, c, /*reuse_a=*/false, /*reuse_b=*/false);
    }

    float* o = out + (size_t)(m0 + 8 * hi) * R_DIM + n0 + lm;
#pragma unroll
    for (int r = 0; r < 8; ++r)
        o[(size_t)r * R_DIM] = c[r];
}

// --------------------------------------------------------------------------
// Init: zero out row 0 (reference never writes it), seed state buffer 0 with
// 0.1, reset barrier counter. Runs AFTER the GEMM (which fills row 0 with pre).
// Must run every launch: workspace is not re-initialized between graph replays.
// --------------------------------------------------------------------------
__global__ void esn_init(float* __restrict__ out, float* __restrict__ state,
                         unsigned* __restrict__ counter)
{
    const int i = blockIdx.x * blockDim.x + threadIdx.x;
    if (i < R_DIM) {
        out[i] = 0.0f;       // row 0 stays zeros
        state[i] = 0.1f;     // x0 = STATE_INIT (buffer 0 read at t=1)
    }
    if (i == 0) *counter = 0u;
}

// --------------------------------------------------------------------------
// Phase 2: persistent scan kernel.
// Block b owns columns [32b, 32b+32): each w_res row touched = one 128B line.
// Wave w covers k in [w*512, (w+1)*512); lane -> column (coalesced).
// State double-buffered: step t reads state[(t-1)&1], writes state[t&1], so a
// single grid barrier per step is race-free.
// --------------------------------------------------------------------------
__global__ __launch_bounds__(TPB) void esn_scan(
    const float* __restrict__ w_res,   // [R_DIM, R_DIM]
    float* __restrict__ out,           // [T_STEPS, R_DIM]; holds pre, gets x_t
    float* __restrict__ state,         // [2, R_DIM]
    unsigned* __restrict__ counter)
{
    __shared__ float xsh[R_DIM];           // 16 KB (of 320 KB/WGP)
    __shared__ float partial[TPB / 32][32];

    const int w    = threadIdx.x >> 5;
    const int lane = threadIdx.x & 31;
    const int j    = blockIdx.x * 32 + lane;   // this thread's output column
    const int k0   = w * (R_DIM / (TPB / 32)); // 512-wide K slice per wave
    const float* wp = w_res + (size_t)k0 * R_DIM + j;

    for (int t = 1; t < T_STEPS; ++t) {
        const float* st_in  = state + (size_t)((t - 1) & 1) * R_DIM;
        float*       st_out = state + (size_t)(t & 1) * R_DIM;

        // Stage x_{t-1} into LDS (coalesced, 16 iterations of 1 KB)
        for (int i = threadIdx.x; i < R_DIM; i += TPB)
            xsh[i] = st_in[i];
        __syncthreads();

        // GEMV slice: 512 rows * 1 float each from L2-resident w_res
        float acc = 0.0f;
#pragma unroll 4
        for (int k = 0; k < R_DIM / (TPB / 32); ++k)
            acc = fmaf(xsh[k0 + k], wp[(size_t)k * R_DIM], acc);

        partial[w][lane] = acc;
        __syncthreads();

        if (w == 0) {
            float y = 0.0f;
#pragma unroll
            for (int q = 0; q < TPB / 32; ++q)
                y += partial[q][lane];
            const float p = out[(size_t)t * R_DIM + j];   // pre_t
            const float x = tanhf(p / y);
            out[(size_t)t * R_DIM + j] = x;
            st_out[j] = x;
        }

        // Grid-wide barrier: release our writes, wait for all NBLK blocks.
        __threadfence();
        __syncthreads();
        if (threadIdx.x == 0) {
            __hip_atomic_fetch_add(counter, 1u, __ATOMIC_RELEASE,
                                   __HIP_MEMORY_SCOPE_AGENT);
            const unsigned target = (unsigned)t * NBLK;
            while (__hip_atomic_load(counter, __ATOMIC_ACQUIRE,
                                     __HIP_MEMORY_SCOPE_AGENT) < target)
                __builtin_amdgcn_s_sleep(2);
        }
        __syncthreads();
    }
}

// --------------------------------------------------------------------------
// Launch: GEMM -> init -> persistent scan, all on `stream`.
// Workspace layout: [2*R_DIM floats state | 1 unsigned counter]
// --------------------------------------------------------------------------
extern "C" void kernel_launch(void* const* d_in, const int* in_sizes, int n_in,
                              void* d_out, int out_size, void* d_ws, size_t ws_size,
                              hipStream_t stream)
{
    const float* train = (const float*)d_in[0];  // [2048, 512]
    const float* w_in  = (const float*)d_in[1];  // [512, 4096]
    const float* w_res = (const float*)d_in[2];  // [4096, 4096]
    float* out = (float*)d_out;                  // [2048, 4096]

    float*    state   = (float*)d_ws;
    unsigned* counter = (unsigned*)((char*)d_ws + 2 * R_DIM * sizeof(float));

    // Phase 1: one wave per 16x16 tile -> 128*256 = 32768 waves, 8 waves/block
    const int gemm_blocks = (T_STEPS / 16) * (R_DIM / 16) / (TPB / 32);
    esn_pre_gemm<<<gemm_blocks, TPB, 0, stream>>>(train, w_in, out);

    esn_init<<<(R_DIM + 255) / 256, 256, 0, stream>>>(out, state, counter);

    // Phase 2: persistent grid (must be fully co-resident for the barrier)
    esn_scan<<<NBLK, TPB, 0, stream>>>(w_res, out, state, counter);
}